// EdgeClassifierGNN_84378927497589
// MI455X (gfx1250) — compile-verified
//
#include <hip/hip_runtime.h>
#include <hip/hip_bf16.h>

// ---------------------------------------------------------------------------
// EdgeClassifierGNN for MI455X (gfx1250, wave32).
// Design: workload is gather/scatter (HBM/L2) bound; matrix FLOPs (~16 GFLOP)
// are tiny, so all GEMMs use full-precision V_WMMA_F32_16X16X4_F32 to match
// the fp32 reference. Edge-MLP tiles are staged through LDS (ds_load_b64 A
// operands). Scatter-add uses fp32 global atomics.
// ---------------------------------------------------------------------------

typedef __attribute__((ext_vector_type(2))) float v2f;
typedef __attribute__((ext_vector_type(8))) float v8f;

__device__ __forceinline__ v8f wmma4(v2f a, v2f b, v8f c) {
  // D = A(16x4, f32) * B(4x16, f32) + C(16x16, f32)
  return __builtin_amdgcn_wmma_f32_16x16x4_f32(
      /*neg_a=*/false, a, /*neg_b=*/false, b,
      /*c_mod=*/(short)0, c, /*reuse_a=*/false, /*reuse_b=*/false);
}

#define HID 64

// ---------------- degree / normalization ----------------
__global__ void zero_k(float* p, int n) {
  int i = blockIdx.x * blockDim.x + threadIdx.x;
  if (i < n) p[i] = 0.0f;
}

__global__ void deg_count_k(const long long* __restrict__ dst, float* deg, int E) {
  int i = blockIdx.x * blockDim.x + threadIdx.x;
  if (i < E) atomicAdd(&deg[dst[i]], 1.0f);
}

__global__ void dinv_k(float* deg, int n) {
  int i = blockIdx.x * blockDim.x + threadIdx.x;
  if (i < n) {
    float d = deg[i] + 1.0f;  // + self loop; always >= 1
    deg[i] = rsqrtf(d);
  }
}

// ---------------- GCN aggregation ----------------
// agg[n] = hpre[n] * dinv[n]^2   (self-loop contribution)
__global__ void agg_init_k(const float* __restrict__ hpre, const float* __restrict__ dinv,
                           float* __restrict__ agg, int N) {
  int i = blockIdx.x * blockDim.x + threadIdx.x;
  if (i < N * HID) {
    int n = i >> 6;
    float di = dinv[n];
    agg[i] = hpre[i] * di * di;
  }
}

// agg[dst] += hpre[src] * dinv[src] * dinv[dst]
__global__ void agg_scatter_k(const float* __restrict__ hpre,
                              const long long* __restrict__ src,
                              const long long* __restrict__ dst,
                              const float* __restrict__ dinv,
                              float* __restrict__ agg, int E) {
  long long i = (long long)blockIdx.x * blockDim.x + threadIdx.x;
  if (i < (long long)E * HID) {
    long long e = i >> 6;
    int f = (int)(i & 63);
    long long s = src[e], d = dst[e];
    float w = dinv[s] * dinv[d];
    atomicAdd(&agg[d * HID + f], hpre[s * HID + f] * w);
  }
}

__global__ void bias_relu_k(float* __restrict__ h, const float* __restrict__ b, int N) {
  int i = blockIdx.x * blockDim.x + threadIdx.x;
  if (i < N * HID) h[i] = fmaxf(h[i] + b[i & 63], 0.0f);
}

// ---------------- dense node GEMM: Y[M,64] = X[M,K] @ W[K,64] ----------------
// One wave per 16-row tile; 4 N-tiles of 16; K-loop of f32 WMMA (K step 4).
__global__ void gemm_node_k(const float* __restrict__ X, const float* __restrict__ W,
                            float* __restrict__ Y, int M, int K) {
  const int lane = threadIdx.x & 31;
  const int mtile = (blockIdx.x * blockDim.x + threadIdx.x) >> 5;
  if (mtile * 16 >= M) return;                     // wave-uniform exit
  const int r = lane & 15, half = lane >> 4;
  int row = mtile * 16 + r;
  if (row >= M) row = M - 1;
  v8f zero = {};
  v8f acc[4] = {zero, zero, zero, zero};
  for (int kb = 0; kb < K; kb += 4) {
    const int k0 = kb + half * 2;                  // A VGPR0/1 hold K=k0,k0+1 (halves K+2)
    v2f a;
    a.x = X[(long long)row * K + k0];
    a.y = X[(long long)row * K + k0 + 1];
#pragma unroll
    for (int t = 0; t < 4; ++t) {
      const int col = t * 16 + r;
      v2f b;
      b.x = W[k0 * HID + col];
      b.y = W[(k0 + 1) * HID + col];
      acc[t] = wmma4(a, b, acc[t]);
    }
  }
#pragma unroll
  for (int t = 0; t < 4; ++t)
#pragma unroll
    for (int v = 0; v < 8; ++v) {
      const int rr = mtile * 16 + v + half * 8;    // C/D layout: M = v + 8*half
      if (rr < M) Y[(long long)rr * HID + t * 16 + r] = acc[t][v];
    }
}

// ---------------- edge MLP ----------------
// out[e,0:2] = relu([h2[src]|h2[dst]|eattr] @ Wm1 + bm1) @ Wm2 + bm2
// One wave per 16-edge tile; gathered A tile staged in LDS; layer-2 uses a
// zero-padded 64x16 B built in registers (only cols 0..1 valid).
__global__ void edge_mlp_k(const float* __restrict__ h2,
                           const long long* __restrict__ src,
                           const long long* __restrict__ dst,
                           const float* __restrict__ eattr,
                           const float* __restrict__ Wm1,
                           const float* __restrict__ bm1,
                           const float* __restrict__ Wm2,
                           const float* __restrict__ bm2,
                           float* __restrict__ out, int E) {
  __shared__ float sA[4][16 * 144];   // per-wave gathered A tile
  __shared__ float sZ[4][16 * 64];    // per-wave layer-1 activations
  const int wId = threadIdx.x >> 5;
  const int lane = threadIdx.x & 31;
  const int tile = blockIdx.x * 4 + wId;
  const int r = lane & 15, half = lane >> 4;

  // ---- gather + stage A tile: row = lane/2, each lane covers 72 columns ----
  {
    const int row = lane >> 1;
    const int seg = lane & 1;
    long long e = (long long)tile * 16 + row;
    if (e >= E) e = E - 1;
    const long long s = src[e], d = dst[e];
    const float* hs = h2 + s * HID;
    const float* hd = h2 + d * HID;
    const float* ea = eattr + e * 16;
    float* a = &sA[wId][row * 144];
    const int c0 = seg * 72;
#pragma unroll
    for (int j = 0; j < 72; ++j) {
      const int c = c0 + j;
      float v;
      if (c < 64)       v = hs[c];
      else if (c < 128) v = hd[c - 64];
      else              v = ea[c - 128];
      a[c] = v;
    }
  }
  asm volatile("s_wait_dscnt 0" ::: "memory");  // LDS writes visible wave-wide

  // ---- layer 1: 16x144 @ 144x64, 36 WMMA K-steps x 4 N-tiles ----
  v8f zero = {};
  v8f acc[4] = {zero, zero, zero, zero};
  const float* arow = &sA[wId][r * 144];
  for (int kb = 0; kb < 144; kb += 4) {
    const int k0 = kb + half * 2;
    const v2f a = *(const v2f*)(arow + k0);       // ds_load_b64
#pragma unroll
    for (int t = 0; t < 4; ++t) {
      const int col = t * 16 + r;
      v2f b;
      b.x = Wm1[k0 * HID + col];
      b.y = Wm1[(k0 + 1) * HID + col];
      acc[t] = wmma4(a, b, acc[t]);
    }
  }

  // bias + ReLU, park z in LDS in C/D layout for re-use as layer-2 A
#pragma unroll
  for (int t = 0; t < 4; ++t) {
    const int col = t * 16 + r;
    const float bias = bm1[col];
#pragma unroll
    for (int v = 0; v < 8; ++v) {
      const int rr = v + half * 8;
      sZ[wId][rr * HID + col] = fmaxf(acc[t][v] + bias, 0.0f);
    }
  }
  asm volatile("s_wait_dscnt 0" ::: "memory");

  // ---- layer 2: z(16x64) @ Wm2_pad(64x16); only output cols 0..1 valid ----
  v8f acc2 = zero;
  const float* zrow = &sZ[wId][r * HID];
  for (int kb = 0; kb < HID; kb += 4) {
    const int k0 = kb + half * 2;
    const v2f a = *(const v2f*)(zrow + k0);
    float bx = Wm2[k0 * 2 + (r & 1)];             // always-in-bounds address
    float by = Wm2[(k0 + 1) * 2 + (r & 1)];
    v2f b;
    b.x = (r < 2) ? bx : 0.0f;
    b.y = (r < 2) ? by : 0.0f;
    acc2 = wmma4(a, b, acc2);
  }
  if (r < 2) {
    const float bias = bm2[r];
#pragma unroll
    for (int v = 0; v < 8; ++v) {
      const long long e = (long long)tile * 16 + v + half * 8;
      if (e < E) out[e * 2 + r] = acc2[v] + bias;
    }
  }
}

// ---------------------------------------------------------------------------
extern "C" void kernel_launch(void* const* d_in, const int* in_sizes, int n_in,
                              void* d_out, int out_size, void* d_ws, size_t ws_size,
                              hipStream_t stream) {
  const float*     x     = (const float*)d_in[0];
  const long long* eidx  = (const long long*)d_in[1];   // int64 [2, E]
  const float*     eattr = (const float*)d_in[2];
  const float*     W1    = (const float*)d_in[3];
  const float*     b1    = (const float*)d_in[4];
  const float*     W2    = (const float*)d_in[5];
  const float*     b2    = (const float*)d_in[6];
  const float*     Wm1   = (const float*)d_in[7];
  const float*     bm1   = (const float*)d_in[8];
  const float*     Wm2   = (const float*)d_in[9];
  const float*     bm2   = (const float*)d_in[10];
  float* out = (float*)d_out;

  const int N = in_sizes[0] / 128;     // 50000
  const int E = in_sizes[1] / 2;       // 800000
  const long long* src = eidx;
  const long long* dstp = eidx + E;

  // workspace layout (floats): dinv[N] | bufA[N*64] | bufB[N*64] | bufC[N*64]
  float* ws = (float*)d_ws;
  float* dinv = ws;
  size_t off = ((size_t)N + 63) & ~(size_t)63;
  float* bufA = ws + off;
  float* bufB = bufA + (size_t)N * HID;
  float* bufC = bufB + (size_t)N * HID;

  const int gN  = (N + 255) / 256;
  const int gE  = (E + 255) / 256;
  const int gNF = (N * HID + 255) / 256;
  const int gEF = (int)(((long long)E * HID + 255) / 256);
  const int tilesN = (N + 15) / 16;
  const int gGemm = (tilesN + 3) / 4;       // 4 waves (tiles) per 128-thread block
  const int tilesE = (E + 15) / 16;
  const int gEdge = (tilesE + 3) / 4;

  // normalization
  zero_k<<<gN, 256, 0, stream>>>(dinv, N);
  deg_count_k<<<gE, 256, 0, stream>>>(dstp, dinv, E);
  dinv_k<<<gN, 256, 0, stream>>>(dinv, N);

  // layer 1: h1 = relu(aggregate(x @ W1) + b1)
  gemm_node_k<<<gGemm, 128, 0, stream>>>(x, W1, bufA, N, 128);
  agg_init_k<<<gNF, 256, 0, stream>>>(bufA, dinv, bufB, N);
  agg_scatter_k<<<gEF, 256, 0, stream>>>(bufA, src, dstp, dinv, bufB, E);
  bias_relu_k<<<gNF, 256, 0, stream>>>(bufB, b1, N);

  // layer 2: h2 = relu(aggregate(h1 @ W2) + b2)
  gemm_node_k<<<gGemm, 128, 0, stream>>>(bufB, W2, bufC, N, HID);
  agg_init_k<<<gNF, 256, 0, stream>>>(bufC, dinv, bufA, N);
  agg_scatter_k<<<gEF, 256, 0, stream>>>(bufC, src, dstp, dinv, bufA, E);
  bias_relu_k<<<gNF, 256, 0, stream>>>(bufA, b2, N);

  // edge MLP
  edge_mlp_k<<<gEdge, 128, 0, stream>>>(bufA, src, dstp, eattr,
                                        Wm1, bm1, Wm2, bm2, out, E);
}